// FreCalc_39831526703557
// MI455X (gfx1250) — compile-verified
//
#include <hip/hip_runtime.h>
#include <math.h>

#define NLAT  128
#define NLON  256
#define LMAX  50
#define MMAX  50
#define NPTS  2048
#define BATCH 2
#define MGRID (NLAT * NLON)   // 32768
#define MMPAD 64              // 50 padded to 4 WMMA tiles

typedef __attribute__((ext_vector_type(2))) float v2f;
typedef __attribute__((ext_vector_type(8))) float v8f;

// ---- workspace layout (float offsets) ----
#define OFF_PHI   0
#define OFF_TH    (OFF_PHI  + BATCH * NPTS)            // 4096
#define OFF_RHO   (OFF_TH   + BATCH * NPTS)            // 8192
#define OFF_F     (OFF_RHO  + BATCH * NPTS)            // 12288
#define OFF_C     (OFF_F    + BATCH * MGRID)           // 77824
#define OFF_FREAL (OFF_C    + NLON * MMPAD)            // 94208
#define OFF_W     (OFF_FREAL+ (BATCH*NLAT) * MMPAD)    // 110592
// total = OFF_W + MMAX*LMAX*NLAT = 430592 floats (~1.72 MB)

__device__ __forceinline__ float pi_f() { return 3.14159265358979323846f; }

// ---------------- 1. cartesian -> spherical ----------------
__global__ void prep_kernel(const float* __restrict__ tgt,
                            float* __restrict__ phi, float* __restrict__ th,
                            float* __restrict__ rho) {
    int i = blockIdx.x * blockDim.x + threadIdx.x;
    if (i >= BATCH * NPTS) return;
    float x = tgt[3 * i + 0], y = tgt[3 * i + 1], z = tgt[3 * i + 2];
    float r = sqrtf(x * x + y * y + z * z);
    phi[i] = atan2f(y, x);
    th[i]  = acosf(z / r) - pi_f();
    rho[i] = r;
}

// ------- 2. Clenshaw-Curtis weights * orthonormal Legendre table -------
// one thread per latitude node k; W[(m*LMAX+l)*NLAT + k] = P_l^m(cos th_k)*w_k
__global__ void sht_w_kernel(float* __restrict__ W) {
    int k = threadIdx.x;
    if (k >= NLAT) return;
    const float PI = pi_f();
    float theta = PI * (float)k / (float)(NLAT - 1);
    float S = 0.f;
    for (int kk = 1; kk <= (NLAT - 1) / 2; ++kk)
        S += 2.0f / (4.0f * kk * kk - 1.0f) * cosf(2.0f * theta * (float)kk);
    float w = (2.0f / (float)(NLAT - 1)) * (1.0f - S);
    if (k == 0 || k == NLAT - 1) w *= 0.5f;

    float x = cosf(theta), s = sinf(theta);
    float pmm = sqrtf(1.0f / (4.0f * PI));       // P[0][0]
    for (int m = 0; m < MMAX; ++m) {
        for (int l = 0; l < m; ++l)
            W[(m * LMAX + l) * NLAT + k] = 0.f;  // P_l^m = 0 for l < m
        float plm2 = pmm;                        // P[m][m]
        W[(m * LMAX + m) * NLAT + k] = plm2 * w;
        if (m + 1 < LMAX) {
            float plm1 = sqrtf(2.f * m + 3.f) * x * plm2;   // P[m][m+1]
            W[(m * LMAX + m + 1) * NLAT + k] = plm1 * w;
            for (int l = m + 2; l < LMAX; ++l) {
                float fl = (float)l, fm = (float)m;
                float a  = sqrtf((4.f * fl * fl - 1.f) / (fl * fl - fm * fm));
                float bb = sqrtf(((fl - 1.f) * (fl - 1.f) - fm * fm) /
                                 (4.f * (fl - 1.f) * (fl - 1.f) - 1.f));
                float pl = a * (x * plm1 - bb * plm2);
                W[(m * LMAX + l) * NLAT + k] = pl * w;
                plm2 = plm1; plm1 = pl;
            }
        }
        float m1 = (float)(m + 1);
        pmm = -sqrtf((2.f * m1 + 1.f) / (2.f * m1)) * s * pmm;  // P[m+1][m+1]
    }
}

// ---------------- 3. cosine (real-rfft) matrix, pre-scaled -------------
__global__ void cmat_kernel(float* __restrict__ C) {
    int i = blockIdx.x * blockDim.x + threadIdx.x;   // i = j*MMPAD + mm
    if (i >= NLON * MMPAD) return;
    int j = i / MMPAD, mm = i % MMPAD;
    const float TWOPI = 6.28318530717958647692f;
    C[i] = cosf(TWOPI * (float)mm * (float)j / (float)NLON) * (TWOPI / (float)NLON);
}

// ---------------- 4. 3-NN + distance-weighted interpolation ------------
// 256 blocks x 256 threads: block = (batch, 256-grid-point slab).
// Candidate points streamed through LDS in 8 tiles of 256.
__global__ void knn_kernel(const float* __restrict__ phi, const float* __restrict__ th,
                           const float* __restrict__ rho, float* __restrict__ f) {
    __shared__ float sp[256], st[256], sr[256];
    int tid = threadIdx.x;
    int b   = blockIdx.x >> 7;                   // 128 blocks per batch
    int m   = ((blockIdx.x & 127) << 8) + tid;   // grid point id in [0, 32768)
    float g0 = (float)(m >> 8) * (pi_f() / (float)NLAT);           // lat comp
    float g1 = (float)((m & 255) - NLAT) * (pi_f() / (float)NLAT); // lon comp

    float d0 = 3.4e38f, d1 = 3.4e38f, d2 = 3.4e38f;
    int   i0 = 0, i1 = 0, i2 = 0;

    for (int t = 0; t < NPTS / 256; ++t) {
        int src = b * NPTS + t * 256 + tid;
        sp[tid] = phi[src]; st[tid] = th[src]; sr[tid] = rho[src];
        __syncthreads();
        #pragma unroll 4
        for (int n = 0; n < 256; ++n) {
            float dp = sp[n] - g0;
            float dt = st[n] - g1;
            float dd = dp * dp + dt * dt;
            int gi = t * 256 + n;
            if (dd < d0)      { d2 = d1; i2 = i1; d1 = d0; i1 = i0; d0 = dd; i0 = gi; }
            else if (dd < d1) { d2 = d1; i2 = i1; d1 = dd; i1 = gi; }
            else if (dd < d2) { d2 = dd; i2 = gi; }
        }
        __syncthreads();
    }
    // recompute exact distances (as reference does) and weight rho by dist/sum
    int base = b * NPTS;
    float p0 = phi[base + i0], t0 = th[base + i0], r0 = rho[base + i0];
    float p1 = phi[base + i1], t1 = th[base + i1], r1 = rho[base + i1];
    float p2 = phi[base + i2], t2 = th[base + i2], r2 = rho[base + i2];
    float e0 = sqrtf((p0 - g0) * (p0 - g0) + (t0 - g1) * (t0 - g1));
    float e1 = sqrtf((p1 - g0) * (p1 - g0) + (t1 - g1) * (t1 - g1));
    float e2 = sqrtf((p2 - g0) * (p2 - g0) + (t2 - g1) * (t2 - g1));
    float inv = 1.0f / (e0 + e1 + e2);
    f[b * MGRID + m] = (r0 * e0 + r1 * e1 + r2 * e2) * inv;
}

// ---------------- 5. DCT matmul via V_WMMA_F32_16X16X4_F32 -------------
// Freal(256 x 64) = f(256 x 256) * C(256 x 64).  One wave32 per 16x16 tile.
// A 16x4 f32 layout: lanes 0-15 hold K={0,1}, lanes 16-31 hold K={2,3}.
// C/D 16x16 f32 layout: VGPR v -> row v (+8 for lanes 16-31), col = lane%16.
__global__ void dct_wmma_kernel(const float* __restrict__ F,
                                const float* __restrict__ C,
                                float* __restrict__ out) {
    int lane = threadIdx.x & 31;
    int wave = blockIdx.x * (blockDim.x >> 5) + (threadIdx.x >> 5);  // 0..63
    int tr = wave >> 2;          // 16 row tiles
    int tc = wave & 3;           // 4 col tiles
    int half = lane >> 4;        // 0: K pair {0,1}, 1: K pair {2,3}
    int rowA = tr * 16 + (lane & 15);
    int colD = tc * 16 + (lane & 15);

    v8f acc = {};
    for (int ko = 0; ko < NLON; ko += 4) {
        v2f a, bm;
        a.x  = F[rowA * NLON + ko + 2 * half + 0];
        a.y  = F[rowA * NLON + ko + 2 * half + 1];
        bm.x = C[(ko + 2 * half + 0) * MMPAD + colD];
        bm.y = C[(ko + 2 * half + 1) * MMPAD + colD];
        acc = __builtin_amdgcn_wmma_f32_16x16x4_f32(
            /*neg_a=*/false, a, /*neg_b=*/false, bm,
            /*c_mod=*/(short)0, acc, /*reuse_a=*/false, /*reuse_b=*/false);
    }
    #pragma unroll
    for (int v = 0; v < 8; ++v)
        out[(tr * 16 + v + 8 * half) * MMPAD + colD] = acc[v];
}

// ---------------- 6. Legendre contraction -> (B, LMAX, MMAX) -----------
__global__ void legendre_kernel(const float* __restrict__ Fr,
                                const float* __restrict__ W,
                                float* __restrict__ out) {
    int i = blockIdx.x * blockDim.x + threadIdx.x;
    if (i >= BATCH * LMAX * MMAX) return;
    int b = i / (LMAX * MMAX);
    int l = (i / MMAX) % LMAX;
    int m = i % MMAX;
    const float* wrow = W + (m * LMAX + l) * NLAT;
    const float* frow = Fr + (b * NLAT) * MMPAD + m;
    float acc = 0.f;
    #pragma unroll 8
    for (int k = 0; k < NLAT; ++k)
        acc = fmaf(frow[k * MMPAD], wrow[k], acc);
    out[i] = acc;
}

extern "C" void kernel_launch(void* const* d_in, const int* in_sizes, int n_in,
                              void* d_out, int out_size, void* d_ws, size_t ws_size,
                              hipStream_t stream) {
    (void)in_sizes; (void)n_in; (void)out_size; (void)ws_size;
    const float* target = (const float*)d_in[0];
    float* ws  = (float*)d_ws;
    float* out = (float*)d_out;

    float* phi   = ws + OFF_PHI;
    float* th    = ws + OFF_TH;
    float* rho   = ws + OFF_RHO;
    float* f     = ws + OFF_F;
    float* Cmat  = ws + OFF_C;
    float* Freal = ws + OFF_FREAL;
    float* W     = ws + OFF_W;

    prep_kernel<<<(BATCH * NPTS + 255) / 256, 256, 0, stream>>>(target, phi, th, rho);
    sht_w_kernel<<<1, NLAT, 0, stream>>>(W);
    cmat_kernel<<<(NLON * MMPAD + 255) / 256, 256, 0, stream>>>(Cmat);
    knn_kernel<<<BATCH * (MGRID / 256), 256, 0, stream>>>(phi, th, rho, f);
    dct_wmma_kernel<<<8, 256, 0, stream>>>(f, Cmat, Freal);   // 64 waves = 64 tiles
    legendre_kernel<<<(BATCH * LMAX * MMAX + 255) / 256, 256, 0, stream>>>(Freal, W, out);
}